// Block_60224031424641
// MI455X (gfx1250) — compile-verified
//
#include <hip/hip_runtime.h>
#include <hip/hip_bf16.h>

typedef float v2f __attribute__((ext_vector_type(2)));
typedef float v8f __attribute__((ext_vector_type(8)));

#define NB   4
#define DIM  64
#define HID  256
#define HH   256
#define WW   256
#define HW   65536
#define EPSV 1e-5f

static __device__ __forceinline__ v8f wmma4(v2f a, v2f b, v8f c) {
    // V_WMMA_F32_16X16X4_F32: D = A(16x4, f32) x B(4x16, f32) + C(16x16, f32)
    return __builtin_amdgcn_wmma_f32_16x16x4_f32(
        /*neg_a=*/false, a, /*neg_b=*/false, b,
        /*c_mod=*/(short)0, c, /*reuse_a=*/false, /*reuse_b=*/false);
}

// ---------------------------------------------------------------------------
// Kernel 1: n1 = BN(x); sum = c55 + c17 + c111 + c211 (all depthwise)
// One block per (n, c, 32x32 tile). LDS tile with halo 10.
// ---------------------------------------------------------------------------
__global__ __launch_bounds__(256) void msca_sum_kernel(
    const float* __restrict__ x,
    const float* __restrict__ g1, const float* __restrict__ b1,
    const float* __restrict__ m1, const float* __restrict__ v1,
    const float* __restrict__ w55,  const float* __restrict__ bb55,
    const float* __restrict__ w17a, const float* __restrict__ b17a,
    const float* __restrict__ w17b, const float* __restrict__ b17b,
    const float* __restrict__ w111a,const float* __restrict__ b111a,
    const float* __restrict__ w111b,const float* __restrict__ b111b,
    const float* __restrict__ w211a,const float* __restrict__ b211a,
    const float* __restrict__ w211b,const float* __restrict__ b211b,
    float* __restrict__ sum)
{
    __shared__ float tile[52][52];
    __shared__ float hbuf[52][32];
    __shared__ float wsh[103];
    __shared__ float bsh[9];

    const int tid = threadIdx.x;
    const int bid = blockIdx.x;
    const int t   = bid & 63;
    const int c   = (bid >> 6) & 63;
    const int n   = bid >> 12;
    const int ty0 = (t >> 3) << 5;
    const int tx0 = (t & 7) << 5;

    // pack per-channel weights: [0..24]=w55 [25..31]=w17a [32..38]=w17b
    // [39..49]=w111a [50..60]=w111b [61..81]=w211a [82..102]=w211b
    if (tid < 25)       wsh[tid] = w55 [c*25 + tid];
    else if (tid < 32)  wsh[tid] = w17a[c*7  + tid-25];
    else if (tid < 39)  wsh[tid] = w17b[c*7  + tid-32];
    else if (tid < 50)  wsh[tid] = w111a[c*11 + tid-39];
    else if (tid < 61)  wsh[tid] = w111b[c*11 + tid-50];
    else if (tid < 82)  wsh[tid] = w211a[c*21 + tid-61];
    else if (tid < 103) wsh[tid] = w211b[c*21 + tid-82];
    if (tid == 103) {
        float sc = g1[c] * rsqrtf(v1[c] + EPSV);
        bsh[0] = sc;
        bsh[1] = b1[c] - m1[c] * sc;
        bsh[2] = bb55[c];
        bsh[3] = b17a[c];  bsh[4] = b17b[c];
        bsh[5] = b111a[c]; bsh[6] = b111b[c];
        bsh[7] = b211a[c]; bsh[8] = b211b[c];
    }
    __syncthreads();

    const float sc = bsh[0], sh = bsh[1];
    const float* xp = x + ((size_t)(n*DIM + c) << 16);
    for (int i = tid; i < 52*52; i += 256) {
        int ly = i / 52, lx = i % 52;
        int gy = ty0 + ly - 10, gx = tx0 + lx - 10;
        float v = 0.f;
        if (gy >= 0 && gy < HH && gx >= 0 && gx < WW)
            v = xp[(gy << 8) + gx] * sc + sh;
        tile[ly][lx] = v;
    }
    __syncthreads();

    float acc[4];
    #pragma unroll
    for (int r = 0; r < 4; ++r) {
        int idx = (r << 8) + tid;
        int oy = idx >> 5, ox = idx & 31;
        float a = bsh[2];
        #pragma unroll
        for (int ky = 0; ky < 5; ++ky)
            #pragma unroll
            for (int kx = 0; kx < 5; ++kx)
                a += wsh[ky*5 + kx] * tile[oy + 8 + ky][ox + 8 + kx];
        acc[r] = a;
    }

    const int whoff[3] = {25, 39, 61};
    const int wvoff[3] = {32, 50, 82};
    const int padk[3]  = {3, 5, 10};
    #pragma unroll
    for (int br = 0; br < 3; ++br) {
        __syncthreads();
        const int p = padk[br];
        const float bh = bsh[3 + br*2];
        // horizontal pass (zero for rows outside the image: zero-padding of
        // the intermediate tensor in the vertical conv)
        for (int i = tid; i < 52*32; i += 256) {
            int row = i >> 5, col = i & 31;
            int gy = ty0 + row - 10;
            float h = 0.f;
            if (gy >= 0 && gy < HH) {
                h = bh;
                for (int j = -p; j <= p; ++j)
                    h += wsh[whoff[br] + j + p] * tile[row][col + 10 + j];
            }
            hbuf[row][col] = h;
        }
        __syncthreads();
        const float bv = bsh[4 + br*2];
        #pragma unroll
        for (int r = 0; r < 4; ++r) {
            int idx = (r << 8) + tid;
            int oy = idx >> 5, ox = idx & 31;
            float a = bv;
            for (int j = -p; j <= p; ++j)
                a += wsh[wvoff[br] + j + p] * hbuf[oy + 10 + j][ox];
            acc[r] += a;
        }
    }

    float* sp = sum + ((size_t)(n*DIM + c) << 16);
    #pragma unroll
    for (int r = 0; r < 4; ++r) {
        int idx = (r << 8) + tid;
        int oy = idx >> 5, ox = idx & 31;
        sp[((ty0 + oy) << 8) + tx0 + ox] = acc[r];
    }
}

// ---------------------------------------------------------------------------
// Kernel 2: mixer = w11 @ sum + b11; x1 = x + ls1 * (mixer * BN1(x))
// WMMA f32 16x16x4, M=64 (out ch), K=64 (in ch), N=pixels.
// Block: 8 waves, 128 pixels; wave: 16 pixels x all 64 out channels.
// ---------------------------------------------------------------------------
__global__ __launch_bounds__(256) void mixer_kernel(
    const float* __restrict__ x, const float* __restrict__ sum,
    const float* __restrict__ w11, const float* __restrict__ b11,
    const float* __restrict__ g1, const float* __restrict__ b1,
    const float* __restrict__ m1, const float* __restrict__ v1,
    const float* __restrict__ ls1, float* __restrict__ x1)
{
    __shared__ float w[DIM*DIM];
    const int tid = threadIdx.x;
    for (int i = tid; i < DIM*DIM; i += 256) w[i] = w11[i];
    __syncthreads();

    const int bid  = blockIdx.x;
    const int n    = bid >> 9;
    const int pblk = (bid & 511) << 7;
    const int wave = tid >> 5, lane = tid & 31;
    const int lane16 = lane & 15;
    const int khalf  = (lane >> 4) << 1;
    const int p0     = pblk + (wave << 4);

    const v8f zero8 = {0.f,0.f,0.f,0.f,0.f,0.f,0.f,0.f};
    v8f acc[4] = {zero8, zero8, zero8, zero8};

    const float* Bbase = sum + ((size_t)n << 22) + p0 + lane16;
    for (int k = 0; k < DIM; k += 4) {
        const int kr = k + khalf;
        v2f b;
        b.x = Bbase[(size_t)kr << 16];
        b.y = Bbase[(size_t)(kr + 1) << 16];
        #pragma unroll
        for (int mt = 0; mt < 4; ++mt) {
            const int row = (mt << 4) + lane16;
            v2f a;
            a.x = w[row*DIM + kr];
            a.y = w[row*DIM + kr + 1];
            acc[mt] = wmma4(a, b, acc[mt]);
        }
    }

    const int p  = p0 + lane16;
    const int hi = (lane >> 4) << 3;
    #pragma unroll
    for (int mt = 0; mt < 4; ++mt) {
        #pragma unroll
        for (int i = 0; i < 8; ++i) {
            const int m = (mt << 4) + hi + i;
            const size_t off = ((size_t)(n*DIM + m) << 16) + p;
            const float xv  = x[off];
            const float scl = g1[m] * rsqrtf(v1[m] + EPSV);
            const float n1v = (xv - m1[m]) * scl + b1[m];
            const float mix = acc[mt][i] + b11[m];
            x1[off] = xv + ls1[m] * (mix * n1v);
        }
    }
}

// ---------------------------------------------------------------------------
// Kernel 3: hdn = fw1 @ BN2(x1) + fb1. M=256, K=64, N=pixels.
// Block: 64 pixels; wave handles 32 out-channels x 64 pixels.
// ---------------------------------------------------------------------------
__global__ __launch_bounds__(256) void ffn1_kernel(
    const float* __restrict__ x1,
    const float* __restrict__ fw1, const float* __restrict__ fb1,
    const float* __restrict__ g2, const float* __restrict__ b2,
    const float* __restrict__ m2, const float* __restrict__ v2,
    float* __restrict__ hdn)
{
    __shared__ float sc2[DIM], sh2[DIM];
    const int tid = threadIdx.x;
    if (tid < DIM) {
        float s = g2[tid] * rsqrtf(v2[tid] + EPSV);
        sc2[tid] = s;
        sh2[tid] = b2[tid] - m2[tid] * s;
    }
    __syncthreads();

    const int bid = blockIdx.x;
    const int n   = bid >> 10;
    const int p0  = (bid & 1023) << 6;
    const int wave = tid >> 5, lane = tid & 31;
    const int lane16 = lane & 15, khalf = (lane >> 4) << 1;

    const v8f zero8 = {0.f,0.f,0.f,0.f,0.f,0.f,0.f,0.f};
    v8f acc[2][4] = {{zero8,zero8,zero8,zero8},{zero8,zero8,zero8,zero8}};

    const float* Bbase = x1 + ((size_t)n << 22) + p0 + lane16;
    for (int k = 0; k < DIM; k += 4) {
        const int kr = k + khalf;
        const float s0 = sc2[kr], h0 = sh2[kr];
        const float s1 = sc2[kr+1], h1 = sh2[kr+1];
        v2f bf[4];
        #pragma unroll
        for (int g = 0; g < 4; ++g) {
            const float* bp = Bbase + (g << 4);
            bf[g].x = bp[(size_t)kr << 16]     * s0 + h0;
            bf[g].y = bp[(size_t)(kr+1) << 16] * s1 + h1;
        }
        #pragma unroll
        for (int mt = 0; mt < 2; ++mt) {
            const int row = ((wave << 1) + mt) * 16 + lane16;
            v2f a;
            a.x = fw1[row*DIM + kr];
            a.y = fw1[row*DIM + kr + 1];
            #pragma unroll
            for (int g = 0; g < 4; ++g)
                acc[mt][g] = wmma4(a, bf[g], acc[mt][g]);
        }
    }

    const int hi = (lane >> 4) << 3;
    #pragma unroll
    for (int mt = 0; mt < 2; ++mt)
        #pragma unroll
        for (int g = 0; g < 4; ++g)
            #pragma unroll
            for (int i = 0; i < 8; ++i) {
                const int hch = ((wave << 1) + mt) * 16 + hi + i;
                const int p = p0 + (g << 4) + lane16;
                hdn[((size_t)(n*HID + hch) << 16) + p] = acc[mt][g][i] + fb1[hch];
            }
}

// ---------------------------------------------------------------------------
// Kernel 4: hdn2 = GELU_erf( dw3x3(hdn) + fbdw )
// ---------------------------------------------------------------------------
__global__ __launch_bounds__(256) void dwgelu_kernel(
    const float* __restrict__ hdn, const float* __restrict__ fdw,
    const float* __restrict__ fbdw, float* __restrict__ hdn2)
{
    __shared__ float t[34][34];
    __shared__ float wk[9];
    __shared__ float bk[1];

    const int tid = threadIdx.x;
    const int bid = blockIdx.x;
    const int tl  = bid & 63;
    const int hc  = (bid >> 6) & 255;
    const int n   = bid >> 14;
    const int ty0 = (tl >> 3) << 5, tx0 = (tl & 7) << 5;

    if (tid < 9) wk[tid] = fdw[hc*9 + tid];
    if (tid == 9) bk[0] = fbdw[hc];

    const float* ip = hdn + ((size_t)(n*HID + hc) << 16);
    for (int i = tid; i < 34*34; i += 256) {
        int ly = i / 34, lx = i % 34;
        int gy = ty0 + ly - 1, gx = tx0 + lx - 1;
        float v = 0.f;
        if (gy >= 0 && gy < HH && gx >= 0 && gx < WW) v = ip[(gy << 8) + gx];
        t[ly][lx] = v;
    }
    __syncthreads();

    float* op = hdn2 + ((size_t)(n*HID + hc) << 16);
    #pragma unroll
    for (int r = 0; r < 4; ++r) {
        int idx = (r << 8) + tid;
        int oy = idx >> 5, ox = idx & 31;
        float a = bk[0];
        #pragma unroll
        for (int ky = 0; ky < 3; ++ky)
            #pragma unroll
            for (int kx = 0; kx < 3; ++kx)
                a += wk[ky*3 + kx] * t[oy + ky][ox + kx];
        float g = 0.5f * a * (1.f + erff(a * 0.70710678118654752f));
        op[((ty0 + oy) << 8) + tx0 + ox] = g;
    }
}

// ---------------------------------------------------------------------------
// Kernel 5: out = x1 + ls2 * (fw2 @ hdn2 + fb2). M=64, K=256, N=pixels.
// Block: 128 pixels; wave: 16 out-ch x 64 pixels.
// ---------------------------------------------------------------------------
__global__ __launch_bounds__(256) void ffn2_kernel(
    const float* __restrict__ hdn2, const float* __restrict__ x1,
    const float* __restrict__ fw2, const float* __restrict__ fb2,
    const float* __restrict__ ls2, float* __restrict__ out)
{
    const int tid = threadIdx.x;
    const int bid = blockIdx.x;
    const int n    = bid >> 9;
    const int pblk = (bid & 511) << 7;
    const int wave = tid >> 5, lane = tid & 31;
    const int lane16 = lane & 15, khalf = (lane >> 4) << 1;
    const int mtile  = wave & 3;
    const int pb     = pblk + ((wave >> 2) << 6);

    const v8f zero8 = {0.f,0.f,0.f,0.f,0.f,0.f,0.f,0.f};
    v8f acc[4] = {zero8, zero8, zero8, zero8};

    const float* Bbase = hdn2 + ((size_t)n << 24) + pb + lane16;
    for (int k = 0; k < HID; k += 4) {
        const int kr = k + khalf;
        v2f bf[4];
        #pragma unroll
        for (int g = 0; g < 4; ++g) {
            bf[g].x = Bbase[((size_t)kr << 16)     + (g << 4)];
            bf[g].y = Bbase[((size_t)(kr+1) << 16) + (g << 4)];
        }
        const int row = (mtile << 4) + lane16;
        v2f a;
        a.x = fw2[row*HID + kr];
        a.y = fw2[row*HID + kr + 1];
        #pragma unroll
        for (int g = 0; g < 4; ++g)
            acc[g] = wmma4(a, bf[g], acc[g]);
    }

    const int hi = (lane >> 4) << 3;
    #pragma unroll
    for (int g = 0; g < 4; ++g)
        #pragma unroll
        for (int i = 0; i < 8; ++i) {
            const int m = (mtile << 4) + hi + i;
            const int p = pb + (g << 4) + lane16;
            const size_t off = ((size_t)n << 22) + ((size_t)m << 16) + p;
            out[off] = x1[off] + ls2[m] * (acc[g][i] + fb2[m]);
        }
}

// ---------------------------------------------------------------------------
extern "C" void kernel_launch(void* const* d_in, const int* in_sizes, int n_in,
                              void* d_out, int out_size, void* d_ws, size_t ws_size,
                              hipStream_t stream) {
    const float* x     = (const float*)d_in[0];
    const float* g1    = (const float*)d_in[1];
    const float* b1    = (const float*)d_in[2];
    const float* m1    = (const float*)d_in[3];
    const float* v1    = (const float*)d_in[4];
    const float* w55   = (const float*)d_in[5];
    const float* bb55  = (const float*)d_in[6];
    const float* w17a  = (const float*)d_in[7];
    const float* b17a  = (const float*)d_in[8];
    const float* w17b  = (const float*)d_in[9];
    const float* b17b  = (const float*)d_in[10];
    const float* w111a = (const float*)d_in[11];
    const float* b111a = (const float*)d_in[12];
    const float* w111b = (const float*)d_in[13];
    const float* b111b = (const float*)d_in[14];
    const float* w211a = (const float*)d_in[15];
    const float* b211a = (const float*)d_in[16];
    const float* w211b = (const float*)d_in[17];
    const float* b211b = (const float*)d_in[18];
    const float* w11   = (const float*)d_in[19];
    const float* b11   = (const float*)d_in[20];
    const float* ls1   = (const float*)d_in[21];
    const float* g2    = (const float*)d_in[22];
    const float* b2    = (const float*)d_in[23];
    const float* m2    = (const float*)d_in[24];
    const float* v2    = (const float*)d_in[25];
    const float* fw1   = (const float*)d_in[26];
    const float* fb1   = (const float*)d_in[27];
    const float* fdw   = (const float*)d_in[28];
    const float* fbdw  = (const float*)d_in[29];
    const float* fw2   = (const float*)d_in[30];
    const float* fb2   = (const float*)d_in[31];
    const float* ls2   = (const float*)d_in[32];

    float* wsf  = (float*)d_ws;
    const size_t SMALL = (size_t)NB * DIM * HW;   // 16,777,216 floats
    const size_t BIG   = (size_t)NB * HID * HW;   // 67,108,864 floats
    float* sum  = wsf;
    float* x1   = wsf + SMALL;
    float* hdn  = wsf + 2*SMALL;
    float* hdn2 = wsf + 2*SMALL + BIG;

    float* out = (float*)d_out;

    msca_sum_kernel<<<NB*DIM*64, 256, 0, stream>>>(
        x, g1, b1, m1, v1, w55, bb55, w17a, b17a, w17b, b17b,
        w111a, b111a, w111b, b111b, w211a, b211a, w211b, b211b, sum);

    mixer_kernel<<<NB*(HW/128), 256, 0, stream>>>(
        x, sum, w11, b11, g1, b1, m1, v1, ls1, x1);

    ffn1_kernel<<<NB*(HW/64), 256, 0, stream>>>(
        x1, fw1, fb1, g2, b2, m2, v2, hdn);

    dwgelu_kernel<<<NB*HID*64, 256, 0, stream>>>(hdn, fdw, fbdw, hdn2);

    ffn2_kernel<<<NB*(HW/128), 256, 0, stream>>>(hdn2, x1, fw2, fb2, ls2, out);
}